// CoordModuleMDS_52329881535141
// MI455X (gfx1250) — compile-verified
//
#include <hip/hip_runtime.h>
#include <hip/hip_bf16.h>
#include <math.h>

// Problem constants (match reference)
#define B_   2
#define L_   768
#define ATM  3
#define N_   (L_ * ATM)      // 2304
#define NCPR 14
#define MDS_ITERS   8
#define POWER_ITERS 16
#define NW    4              // wave32s per block
#define BLOCK (NW * 32)
#define NTILES (N_ / 16)     // 144

typedef __attribute__((ext_vector_type(2))) float v2f;
typedef __attribute__((ext_vector_type(8))) float v8f;
typedef __attribute__((ext_vector_type(4))) int v4i;

__device__ __forceinline__ v8f wmma_f32_16x16x4(v2f a, v2f b, v8f c) {
  // D(16x16) = A(16x4) * B(4x16) + C ; f32 throughout
  return __builtin_amdgcn_wmma_f32_16x16x4_f32(false, a, false, b, (short)0, c,
                                               false, false);
}

// Async global->LDS 16-byte copy (CDNA5 GLOBAL_LOAD_ASYNC_TO_LDS_B128), with
// portable fallback. Each calling lane copies 16 bytes to its own LDS address.
// Builtin prototype (from probe): (as1 v4i* gsrc, as3 v4i* ldst, imm, imm cpol)
__device__ __forceinline__ void async_copy_b128(const float4* g, float4* l) {
#if __has_builtin(__builtin_amdgcn_global_load_async_to_lds_b128)
  typedef __attribute__((address_space(1))) v4i as1_v4i;
  typedef __attribute__((address_space(3))) v4i as3_v4i;
  __builtin_amdgcn_global_load_async_to_lds_b128((as1_v4i*)g, (as3_v4i*)l, 0, 0);
#else
  *l = *g;
#endif
}

__device__ __forceinline__ void wait_async0() {
#if __has_builtin(__builtin_amdgcn_s_wait_asynccnt)
  __builtin_amdgcn_s_wait_asynccnt(0);
#elif defined(__HIP_DEVICE_COMPILE__)
  asm volatile("s_wait_asynccnt 0x0" ::: "memory");
#endif
}

// Branch-free select of B-fragment value from a staged float4; cols >= 4 -> 0.
__device__ __forceinline__ float bsel(float4 P, float w3, int c) {
  float r = (c == 0) ? P.x : 0.0f;
  r = (c == 1) ? P.y : r;
  r = (c == 2) ? P.z : r;
  r = (c == 3) ? w3 : r;
  return r;
}

// ---------------------------------------------------------------------------
// Kernel: squared distances of row 0 (Gram matrix ingredient)
__global__ void k_sq0(const float* __restrict__ dp, float* __restrict__ sq0) {
  int t = blockIdx.x * blockDim.x + threadIdx.x;
  if (t < B_ * N_) {
    int b = t / N_, j = t % N_;
    float d = dp[(((size_t)b * N_ + 0) * N_ + j) * 2];
    sq0[t] = d * d;
  }
}

// Kernel: deterministic init of the 3-column subspace, padded layout:
// V4[i] = {v0, v1, v2, sq0[i]}
__global__ void k_initv(const float* __restrict__ sq0, float4* __restrict__ V4) {
  int t = blockIdx.x * blockDim.x + threadIdx.x;
  if (t < B_ * N_) {
    int i = t % N_;
    int b = t / N_;
    float4 v;
    v.x = sinf(0.0137f * (float)(i + 1) * 1.0f + 0.303f * (float)b);
    v.y = sinf(0.0137f * (float)(i + 1) * 2.0f + 0.71f + 0.303f * (float)b);
    v.z = sinf(0.0137f * (float)(i + 1) * 3.0f + 1.42f + 0.303f * (float)b);
    v.w = sq0[t];
    V4[t] = v;
  }
}

// ---------------------------------------------------------------------------
// Kernel: W = M * V via V_WMMA_F32_16X16X4_F32 (block power iteration step).
// M_ij = 0.5*(sq0[i] + sq0[j] - d_ij^2) built on the fly; V and sq0 arrive
// together in the staged float4 (w component = sq0).
__global__ void k_mv(const float* __restrict__ dp, const float4* __restrict__ V4,
                     float* __restrict__ W) {
  __shared__ float4 vst[N_];
  __shared__ float red[NW][16][4];

  int blk = blockIdx.x;
  int b = blk / NTILES;
  int i0 = (blk % NTILES) * 16;
  int tid = threadIdx.x, lane = tid & 31, wv = tid >> 5;

  for (int t = tid; t < N_; t += BLOCK)
    async_copy_b128(&V4[(size_t)b * N_ + t], &vst[t]);
  wait_async0();
  __syncthreads();

  int mrow = lane & 15;        // A row within tile; also B column
  int half = lane >> 4;        // K half: lanes<16 -> K{0,1}, lanes>=16 -> K{2,3}
  int i = i0 + mrow;
  const float* rowp = dp + ((size_t)b * N_ + i) * N_ * 2;
  float sqi = vst[i].w;

  v8f acc = {0.f, 0.f, 0.f, 0.f, 0.f, 0.f, 0.f, 0.f};
  for (int k = wv * 4; k < N_; k += NW * 4) {
    int jb = k + half * 2;
    float4 q = *(const float4*)(rowp + (size_t)jb * 2);  // d_j, std_j, d_j1, std_j1
    __builtin_prefetch(rowp + (size_t)(jb + NW * 4) * 2, 0, 3);

    float4 P0 = vst[jb];
    float4 P1 = vst[jb + 1];
    v2f a, bf;
    a.x = 0.5f * (sqi + P0.w - q.x * q.x);
    a.y = 0.5f * (sqi + P1.w - q.z * q.z);
    bf.x = bsel(P0, 0.0f, mrow);   // B columns: [v0 v1 v2 0]
    bf.y = bsel(P1, 0.0f, mrow);
    acc = wmma_f32_16x16x4(a, bf, acc);
  }

  // D layout: lane<16 -> col=lane, rows v; lane>=16 -> col=lane-16, rows v+8
  if (mrow < 4) {
    int rb = half * 8;
#pragma unroll
    for (int v = 0; v < 8; ++v) red[wv][rb + v][mrow] = acc[v];
  }
  __syncthreads();
  if (tid < 64) {
    int row = tid >> 2, c = tid & 3;
    float s = 0.f;
    for (int w2 = 0; w2 < NW; ++w2) s += red[w2][row][c];
    red[0][row][c] = s;
  }
  __syncthreads();
  if (tid < 48) {
    int row = tid / 3, d = tid % 3;
    W[((size_t)b * N_ + (i0 + row)) * 3 + d] = red[0][row][d];
  }
}

// ---------------------------------------------------------------------------
// Kernel: orthonormalize W (3x3 Cholesky QR) -> V4; on final pass also emit
// coords4 = {Q_k * sqrt(|rayleigh_k|), 1}, columns sorted by |lambda| desc.
__global__ void k_orth(float4* __restrict__ V4, const float* __restrict__ W,
                       const float* __restrict__ sq0, float4* __restrict__ coords4,
                       int finalFlag) {
  const int T = 256;
  __shared__ float sred[9 * T];
  __shared__ float Rm[6];
  __shared__ float scl[3];
  __shared__ int perm[3];

  int b = blockIdx.x, tid = threadIdx.x;
  const float* Wb = W + (size_t)b * N_ * 3;
  const float4* Vb = V4 + (size_t)b * N_;

  float p[9] = {0.f, 0.f, 0.f, 0.f, 0.f, 0.f, 0.f, 0.f, 0.f};
  for (int i = tid; i < N_; i += T) {
    float w0 = Wb[i * 3], w1 = Wb[i * 3 + 1], w2 = Wb[i * 3 + 2];
    float4 v = Vb[i];
    p[0] += w0 * w0; p[1] += w0 * w1; p[2] += w0 * w2;
    p[3] += w1 * w1; p[4] += w1 * w2; p[5] += w2 * w2;
    p[6] += v.x * w0; p[7] += v.y * w1; p[8] += v.z * w2;
  }
  for (int q = 0; q < 9; ++q) sred[q * T + tid] = p[q];
  __syncthreads();
  for (int s = T / 2; s > 0; s >>= 1) {
    if (tid < s)
      for (int q = 0; q < 9; ++q) sred[q * T + tid] += sred[q * T + tid + s];
    __syncthreads();
  }
  if (tid == 0) {
    float g00 = sred[0 * T], g01 = sred[1 * T], g02 = sred[2 * T];
    float g11 = sred[3 * T], g12 = sred[4 * T], g22 = sred[5 * T];
    float lam[3] = {sred[6 * T], sred[7 * T], sred[8 * T]};
    float r00 = sqrtf(fmaxf(g00, 1e-20f));
    float r01 = g01 / r00, r02 = g02 / r00;
    float r11 = sqrtf(fmaxf(g11 - r01 * r01, 1e-20f));
    float r12 = (g12 - r01 * r02) / r11;
    float r22 = sqrtf(fmaxf(g22 - r02 * r02 - r12 * r12, 1e-20f));
    Rm[0] = r00; Rm[1] = r01; Rm[2] = r02; Rm[3] = r11; Rm[4] = r12; Rm[5] = r22;
    int pm[3] = {0, 1, 2};
    float al[3] = {fabsf(lam[0]), fabsf(lam[1]), fabsf(lam[2])};
    if (al[pm[0]] < al[pm[1]]) { int t = pm[0]; pm[0] = pm[1]; pm[1] = t; }
    if (al[pm[0]] < al[pm[2]]) { int t = pm[0]; pm[0] = pm[2]; pm[2] = t; }
    if (al[pm[1]] < al[pm[2]]) { int t = pm[1]; pm[1] = pm[2]; pm[2] = t; }
    for (int k = 0; k < 3; ++k) { perm[k] = pm[k]; scl[k] = sqrtf(al[pm[k]]); }
  }
  __syncthreads();

  float r00 = Rm[0], r01 = Rm[1], r02 = Rm[2], r11 = Rm[3], r12 = Rm[4], r22 = Rm[5];
  for (int i = tid; i < N_; i += T) {
    float w0 = Wb[i * 3], w1 = Wb[i * 3 + 1], w2 = Wb[i * 3 + 2];
    float q0 = w0 / r00;
    float q1 = (w1 - r01 * q0) / r11;
    float q2 = (w2 - r02 * q0 - r12 * q1) / r22;
    float qv[3] = {q0, q1, q2};
    float4 vout;
    vout.x = q0; vout.y = q1; vout.z = q2; vout.w = sq0[b * N_ + i];
    V4[(size_t)b * N_ + i] = vout;
    if (finalFlag) {
      float4 c;
      c.x = qv[perm[0]] * scl[0];
      c.y = qv[perm[1]] * scl[1];
      c.z = qv[perm[2]] * scl[2];
      c.w = 1.0f;  // rowsum column for the Guttman WMMA B operand
      coords4[(size_t)b * N_ + i] = c;
    }
  }
}

// ---------------------------------------------------------------------------
// Kernel: one Guttman transform iteration via WMMA.
// B operand columns = [cx, cy, cz, 1] -> D gives both ratio@c and rowsum(ratio).
__global__ void k_guttman(const float* __restrict__ dp,
                          const unsigned char* __restrict__ bbm,
                          const float4* __restrict__ cin,
                          float4* __restrict__ cout) {
  __shared__ float4 bst[N_];
  __shared__ float red[NW][16][4];

  int blk = blockIdx.x;
  int b = blk / NTILES;
  int i0 = (blk % NTILES) * 16;
  int tid = threadIdx.x, lane = tid & 31, wv = tid >> 5;

  for (int t = tid; t < N_; t += BLOCK)
    async_copy_b128(&cin[(size_t)b * N_ + t], &bst[t]);
  wait_async0();
  __syncthreads();

  int mrow = lane & 15;
  int half = lane >> 4;
  int i = i0 + mrow;
  const float* rowp = dp + ((size_t)b * N_ + i) * N_ * 2;
  const unsigned char* mp = bbm + ((size_t)b * N_ + i) * N_;
  float4 ci = bst[i];

  v8f acc = {0.f, 0.f, 0.f, 0.f, 0.f, 0.f, 0.f, 0.f};
  for (int k = wv * 4; k < N_; k += NW * 4) {
    int jb = k + half * 2;
    float4 q = *(const float4*)(rowp + (size_t)jb * 2);  // d_j, std_j, d_j1, std_j1
    __builtin_prefetch(rowp + (size_t)(jb + NW * 4) * 2, 0, 3);
    unsigned int mm = *(const unsigned short*)(mp + jb);  // 2 mask bytes

    float4 P0 = bst[jb];
    float4 P1 = bst[jb + 1];
    v2f a, bf;
    {
      float w = (mm & 0xffu) ? (1.0f / (1.0f + q.y)) : 0.0f;
      float dx = ci.x - P0.x, dy = ci.y - P0.y, dz = ci.z - P0.z;
      float dist = sqrtf(dx * dx + dy * dy + dz * dz + 1e-12f);
      a.x = w * q.x / dist;
    }
    {
      float w = (mm >> 8) ? (1.0f / (1.0f + q.w)) : 0.0f;
      float dx = ci.x - P1.x, dy = ci.y - P1.y, dz = ci.z - P1.z;
      float dist = sqrtf(dx * dx + dy * dy + dz * dz + 1e-12f);
      a.y = w * q.z / dist;
    }
    bf.x = bsel(P0, P0.w, mrow);  // B columns: [cx cy cz 1]
    bf.y = bsel(P1, P1.w, mrow);
    acc = wmma_f32_16x16x4(a, bf, acc);
  }

  if (mrow < 4) {
    int rb = half * 8;
#pragma unroll
    for (int v = 0; v < 8; ++v) red[wv][rb + v][mrow] = acc[v];
  }
  __syncthreads();
  if (tid < 64) {
    int row = tid >> 2, c = tid & 3;
    float s = 0.f;
    for (int w2 = 0; w2 < NW; ++w2) s += red[w2][row][c];
    red[0][row][c] = s;
  }
  __syncthreads();
  if (tid < 16) {
    int row = tid;
    int ig = i0 + row;
    float rs = red[0][row][3];  // rowsum(ratio)
    const float inv_n = 1.0f / (float)N_;
    float4 c = bst[ig];
    float4 o;
    o.x = (rs * c.x - red[0][row][0]) * inv_n;
    o.y = (rs * c.y - red[0][row][1]) * inv_n;
    o.z = (rs * c.z - red[0][row][2]) * inv_n;
    o.w = 1.0f;
    cout[(size_t)b * N_ + ig] = o;
  }
}

// ---------------------------------------------------------------------------
// Kernel: dihedral sign vote -> z flip factor per batch
__global__ void k_flip(const float4* __restrict__ c, float* __restrict__ flipz) {
  __shared__ int cnt[256];
  int b = blockIdx.x, tid = threadIdx.x;
  const float4* base = c + (size_t)b * N_;
  int local = 0;
  for (int r = tid; r < L_ - 1; r += blockDim.x) {
    float4 p1 = base[r * 3 + 2];        // C of res r
    float4 p2 = base[(r + 1) * 3 + 0];  // N of res r+1
    float4 p3 = base[(r + 1) * 3 + 1];  // CA of res r+1
    float4 p4 = base[(r + 1) * 3 + 2];  // C of res r+1
    float u1x = p2.x - p1.x, u1y = p2.y - p1.y, u1z = p2.z - p1.z;
    float u2x = p3.x - p2.x, u2y = p3.y - p2.y, u2z = p3.z - p2.z;
    float u3x = p4.x - p3.x, u3y = p4.y - p3.y, u3z = p4.z - p3.z;
    float cx = u2y * u3z - u2z * u3y;
    float cy = u2z * u3x - u2x * u3z;
    float cz = u2x * u3y - u2y * u3x;
    float y = u1x * cx + u1y * cy + u1z * cz;  // sign(phi) == sign(y), |u2|>0
    local += (y < 0.0f) ? 1 : 0;
  }
  cnt[tid] = local;
  __syncthreads();
  for (int s = blockDim.x / 2; s > 0; s >>= 1) {
    if (tid < s) cnt[tid] += cnt[tid + s];
    __syncthreads();
  }
  if (tid == 0) {
    float mean = (float)cnt[0] / (float)(L_ - 1);
    flipz[b] = (mean < 0.5f) ? -1.0f : 1.0f;
  }
}

// Kernel: expand to 14 atoms per residue with cloud mask + z flip
__global__ void k_out(const float4* __restrict__ c,
                      const unsigned char* __restrict__ cm,
                      const float* __restrict__ flipz, float* __restrict__ out) {
  int t = blockIdx.x * blockDim.x + threadIdx.x;
  const int total = B_ * L_ * NCPR;
  if (t >= total) return;
  int a = t % NCPR;
  int l = (t / NCPR) % L_;
  int b = t / (NCPR * L_);
  float x = 0.f, y = 0.f, z = 0.f;
  if (cm[t]) {
    int src = (a < ATM) ? a : 1;  // c_beta = 1 for num_atoms == 3
    float4 p = c[(size_t)b * N_ + (size_t)l * ATM + src];
    x = p.x; y = p.y; z = p.z * flipz[b];
  }
  out[(size_t)t * 3 + 0] = x;
  out[(size_t)t * 3 + 1] = y;
  out[(size_t)t * 3 + 2] = z;
}

// ---------------------------------------------------------------------------
extern "C" void kernel_launch(void* const* d_in, const int* in_sizes, int n_in,
                              void* d_out, int out_size, void* d_ws, size_t ws_size,
                              hipStream_t stream) {
  (void)in_sizes; (void)n_in; (void)out_size; (void)ws_size;
  const float* dp = (const float*)d_in[1];                  // distance_pred (B,N,N,2)
  const unsigned char* cm = (const unsigned char*)d_in[6];  // cloud_mask (B,L,14)
  const unsigned char* bbm = (const unsigned char*)d_in[7]; // bb_flat_mask_crossed
  float* out = (float*)d_out;

  float* ws = (float*)d_ws;
  float* sq0 = ws;                              // B*N floats
  float4* V4 = (float4*)(ws + B_ * N_);         // B*N float4
  float* W = (float*)(V4 + B_ * N_);            // B*N*3 floats
  float4* cA = (float4*)(W + B_ * N_ * 3);      // B*N float4
  float4* cB = cA + B_ * N_;                    // B*N float4
  float* flipz = (float*)(cB + B_ * N_);        // B floats

  k_sq0<<<(B_ * N_ + 255) / 256, 256, 0, stream>>>(dp, sq0);
  k_initv<<<(B_ * N_ + 255) / 256, 256, 0, stream>>>(sq0, V4);

  for (int it = 0; it < POWER_ITERS; ++it) {
    k_mv<<<B_ * NTILES, BLOCK, 0, stream>>>(dp, V4, W);
    k_orth<<<B_, 256, 0, stream>>>(V4, W, sq0, cA, (it == POWER_ITERS - 1) ? 1 : 0);
  }

  float4* ci = cA;
  float4* co = cB;
  for (int it = 0; it < MDS_ITERS; ++it) {
    k_guttman<<<B_ * NTILES, BLOCK, 0, stream>>>(dp, bbm, ci, co);
    float4* t = ci; ci = co; co = t;
  }
  // MDS_ITERS is even -> result back in cA (== ci)

  k_flip<<<B_, 256, 0, stream>>>(ci, flipz);
  k_out<<<(B_ * L_ * NCPR + 255) / 256, 256, 0, stream>>>(ci, cm, flipz, out);
}